// GPES_57939108823834
// MI455X (gfx1250) — compile-verified
//
#include <hip/hip_runtime.h>
#include <math.h>

// ---------------------------------------------------------------------------
// Problem constants (match reference)
// ---------------------------------------------------------------------------
#define BATCH   256
#define IN_DIM  159
#define KPAD    160     // IN_DIM padded to multiple of 4
#define MDIM    93
#define MATEL   2976
#define NCLS    10
#define SIGDIM  340
#define KPATH   64
#define LPATH   63
#define NSEG    62      // LPATH-1
#define MPAD    96      // MDIM padded to multiple of 16

typedef float v2f __attribute__((ext_vector_type(2)));
typedef float v8f __attribute__((ext_vector_type(8)));

// f32 WMMA: D(16x16) = A(16x4) * B(4x16) + C   -- full fp32 precision
__device__ __forceinline__ v8f wmma4(v2f a, v2f b, v8f c) {
    return __builtin_amdgcn_wmma_f32_16x16x4_f32(
        /*neg_a=*/false, a, /*neg_b=*/false, b,
        /*c_mod=*/(short)0, c, /*reuse_a=*/false, /*reuse_b=*/false);
}

// ---------------------------------------------------------------------------
// Helpers: zero (grid-stride) / pad copies (straight-line, one element per thread)
// ---------------------------------------------------------------------------
__global__ void zero_kernel(float* __restrict__ p, int n) {
    for (int i = blockIdx.x * blockDim.x + threadIdx.x; i < n;
         i += gridDim.x * blockDim.x)
        p[i] = 0.0f;
}

// dst[r][c] (r<Npad, c<KPAD) = (r<Nsrc && c<IN_DIM) ? src[r][c] : 0
__global__ void pad_rows_k(const float* __restrict__ src, float* __restrict__ dst,
                           int Nsrc, int Npad) {
    const int i = blockIdx.x * blockDim.x + threadIdx.x;
    if (i >= Npad * KPAD) return;
    const int r = i / KPAD, c = i - r * KPAD;
    dst[i] = (r < Nsrc && c < IN_DIM) ? src[r * IN_DIM + c] : 0.0f;
}

// epsT[b][n][r] = (r<93) ? eps[b][r][n] : 0     (transpose so (k,k+1) contiguous)
__global__ void pad_eps_T(const float* __restrict__ eps, float* __restrict__ epsT) {
    const int i = blockIdx.x * blockDim.x + threadIdx.x;
    if (i >= BATCH * KPATH * MPAD) return;
    const int b = i / (KPATH * MPAD);
    const int rem = i - b * (KPATH * MPAD);
    const int n = rem / MPAD, r = rem - n * MPAD;
    epsT[i] = (r < MDIM) ? eps[(b * MDIM + r) * KPATH + n] : 0.0f;
}

// ---------------------------------------------------------------------------
// Y[r][c] = sum_k X[r][k]*W[c][k] + bias[c]
// X: [256][160] padded, W: [Npad][160] padded (zero rows/cols).
// grid = (ceil(coltiles/4), 16 row tiles), block = 128 (4 waves, 1 tile each).
// Inner loop: 2x global_load_b64 + v_wmma, no conditionals.
// ---------------------------------------------------------------------------
__global__ void gemm_xwT(const float* __restrict__ X, const float* __restrict__ W,
                         const float* __restrict__ bias, float* __restrict__ Y,
                         int NoutReal, int NoutStride, int coltiles) {
    const int wid  = threadIdx.x >> 5;
    const int lane = threadIdx.x & 31;
    const int ct   = blockIdx.x * 4 + wid;
    if (ct >= coltiles) return;                      // wave-uniform exit
    const int r0 = blockIdx.y * 16, c0 = ct * 16;
    const int m  = lane & 15, hi = lane >> 4;
    const float* Xr = X + (r0 + m) * KPAD;
    const float* Wr = W + (c0 + m) * KPAD;
    v8f acc = {};
    for (int k0 = 0; k0 < KPAD; k0 += 4) {
        const int ka = k0 + 2 * hi;
        const v2f a = *reinterpret_cast<const v2f*>(Xr + ka);
        const v2f b = *reinterpret_cast<const v2f*>(Wr + ka);
        acc = wmma4(a, b, acc);
    }
    const int ocol = c0 + m;
    if (ocol < NoutReal) {
        const float bv = bias[ocol];
#pragma unroll
        for (int j = 0; j < 8; ++j)                  // C layout: vgpr j -> M=j / j+8
            Y[(r0 + j + 8 * hi) * NoutStride + ocol] = acc[j] + bv;
    }
}

// ---------------------------------------------------------------------------
// Scatter banded sqrt-covariance into dense padded M[256][96][96]
// ---------------------------------------------------------------------------
__global__ void scatter_cov(const float* __restrict__ cov,
                            const int* __restrict__ xi, const int* __restrict__ yi,
                            float* __restrict__ Mbuf) {
    const int t = blockIdx.x * blockDim.x + threadIdx.x;
    if (t >= BATCH * MATEL) return;
    const int b = t / MATEL, i = t - b * MATEL;
    Mbuf[b * (MPAD * MPAD) + xi[i] * MPAD + yi[i]] = cov[b * MATEL + i];
}

// ---------------------------------------------------------------------------
// newV[b] = M[b](96x96) @ eps[b](96x64) + mean[b]   (all padded, no guards)
// block = 192 threads = 6 waves = one batch; wave w owns 16 rows x 64 cols.
// Per k-step: 1 A-fragment load shared by 4 accumulators (4 B loads, 4 WMMA).
// ---------------------------------------------------------------------------
__global__ __launch_bounds__(192)
void gemm_batched(const float* __restrict__ Mbuf, const float* __restrict__ epsT,
                  const float* __restrict__ mean, float* __restrict__ newV) {
    const int wid  = threadIdx.x >> 5;               // row tile 0..5
    const int lane = threadIdx.x & 31;
    const int b  = blockIdx.x;
    const int r0 = wid * 16;
    const int m  = lane & 15, hi = lane >> 4;
    const float* Ar = Mbuf + b * (MPAD * MPAD) + (r0 + m) * MPAD;
    const float* Eb = epsT + b * (KPATH * MPAD);     // [n][k] layout
    v8f acc[4] = {{}, {}, {}, {}};
    for (int k0 = 0; k0 < MPAD; k0 += 4) {
        const int ka = k0 + 2 * hi;
        const v2f a = *reinterpret_cast<const v2f*>(Ar + ka);
#pragma unroll
        for (int ct = 0; ct < 4; ++ct) {
            const v2f bv = *reinterpret_cast<const v2f*>(Eb + (ct * 16 + m) * MPAD + ka);
            acc[ct] = wmma4(a, bv, acc[ct]);
        }
    }
#pragma unroll
    for (int j = 0; j < 8; ++j) {
        const int row = r0 + j + 8 * hi;
        const float mv = mean[b * MPAD + row];       // mean pre-zeroed incl. pad rows
#pragma unroll
        for (int ct = 0; ct < 4; ++ct)
            newV[(b * MPAD + row) * KPATH + ct * 16 + m] = acc[ct][j] + mv;
    }
}

// ---------------------------------------------------------------------------
// Signature kernel: one block per batch row b, 256 threads = 64 paths x 4 lanes.
// Lane q of a path-group owns the a=q slice of S1/S2/S3/S4 (1+4+16+64 = 85 f32).
// Segment exps are dx^{(x)k}/k!, so each step is a Horner-form FMA cascade and
// the whole scan is lane-local (dx broadcast through LDS as float4).
// ---------------------------------------------------------------------------
__global__ __launch_bounds__(256)
void sig_kernel(const float* __restrict__ x, const float* __restrict__ newV,
                const float* __restrict__ Wfin, const float* __restrict__ bfin,
                float* __restrict__ out) {
    __shared__ float smem[KPATH * NSEG * 4];         // 63488 B, reused for reduce
    __shared__ float logits[NCLS];

    const int b = blockIdx.x;
    const int t = threadIdx.x;
    const int k = t >> 2;                            // path id within batch row
    const int q = t & 3;                             // owned channel / slice
    const float* xb = x + b * IN_DIM;

    // ---- Phase A: build dx into LDS (lane q supplies channel q) ----
    auto fval = [&](int i) -> float {
        const int o = (i & 1) ? (32 + (i >> 1)) : (i >> 1);   // ORDER[i]
        if (q == 3) return xb[96 + o];                        // time channel
        if (o < 32) return xb[o * 3 + q];                     // old point
        return newV[((b * MPAD) + (o - 32) * 3 + q) * KPATH + k]; // sampled point
    };
    float prev = fval(0);
    for (int i = 1; i < LPATH; ++i) {
        const float cur = fval(i);
        smem[(k * NSEG + (i - 1)) * 4 + q] = cur - prev;
        prev = cur;
    }
    __syncthreads();

    // ---- Phase B: Chen scan, levels 1..4 ----
    float S1 = 0.0f, S2[4] = {}, S3[16] = {}, S4[64] = {};
    for (int i = 0; i < NSEG; ++i) {
        const float4 d4 = *reinterpret_cast<const float4*>(&smem[(k * NSEG + i) * 4]);
        const float d[4] = {d4.x, d4.y, d4.z, d4.w};
        const float dq = (q == 0) ? d4.x : (q == 1) ? d4.y : (q == 2) ? d4.z : d4.w;
        const float base4 = S1 * (1.0f / 6.0f) + dq * (1.0f / 24.0f);
        const float base3 = S1 * 0.5f          + dq * (1.0f / 6.0f);
        const float base2 = S1                 + dq * 0.5f;
#pragma unroll
        for (int bb = 0; bb < 4; ++bb) {
            const float t4 = 0.5f * S2[bb] + d[bb] * base4;
            const float t3 =        S2[bb] + d[bb] * base3;
#pragma unroll
            for (int c = 0; c < 4; ++c) {
                const float u4 = S3[bb * 4 + c] + d[c] * t4;
#pragma unroll
                for (int e = 0; e < 4; ++e)
                    S4[bb * 16 + c * 4 + e] += d[e] * u4;
                S3[bb * 4 + c] += d[c] * t3;
            }
            S2[bb] += d[bb] * base2;
        }
        S1 += dq;
    }

    // ---- Phase C: per-level squared norms (reduce over the 4 slice-lanes) ----
    float n1 = S1 * S1, n2 = 0.0f, n3 = 0.0f, n4 = 0.0f;
#pragma unroll
    for (int j = 0; j < 4; ++j)  n2 += S2[j] * S2[j];
#pragma unroll
    for (int j = 0; j < 16; ++j) n3 += S3[j] * S3[j];
#pragma unroll
    for (int j = 0; j < 64; ++j) n4 += S4[j] * S4[j];
#pragma unroll
    for (int mask = 1; mask <= 2; mask <<= 1) {
        n1 += __shfl_xor(n1, mask, 32);
        n2 += __shfl_xor(n2, mask, 32);
        n3 += __shfl_xor(n3, mask, 32);
        n4 += __shfl_xor(n4, mask, 32);
    }

    // phi with C=4, A=1:  psi = x (x<=4) else 8 - 16/x ; bisection for lambda
    const float norm2 = 1.0f + n1 + n2 + n3 + n4;
    const float psi = (norm2 <= 4.0f) ? norm2 : (8.0f - 16.0f / norm2);
    float lo = 0.0f, hi = 1.0f;
    for (int it = 0; it < 40; ++it) {
        const float mid = 0.5f * (lo + hi);
        const float m2 = mid * mid, m4 = m2 * m2;
        const float val = 1.0f + m2 * n1 + m4 * n2 + m4 * m2 * n3 + m4 * m4 * n4;
        if (val > psi) hi = mid; else lo = mid;
    }
    const float lam = 0.5f * (lo + hi);
    const float l1 = lam, l2 = lam * lam, l3 = l2 * lam, l4 = l2 * l2;

    // ---- Phase D: sum normalized signatures over K=64 paths (LDS) ----
    __syncthreads();                                  // done with dx region
    float* red = smem;                                // reuse: 340 accumulators
    for (int s = t; s < SIGDIM; s += 256) red[s] = 0.0f;
    __syncthreads();
    atomicAdd(&red[q], S1 * l1);
#pragma unroll
    for (int j = 0; j < 4; ++j)  atomicAdd(&red[4  + q * 4  + j], S2[j] * l2);
#pragma unroll
    for (int j = 0; j < 16; ++j) atomicAdd(&red[20 + q * 16 + j], S3[j] * l3);
#pragma unroll
    for (int j = 0; j < 64; ++j) atomicAdd(&red[84 + q * 64 + j], S4[j] * l4);
    __syncthreads();

    // ---- Phase E: final linear (mean over K) + log_softmax ----
    if (t < NCLS) {
        float acc = 0.0f;
        for (int s = 0; s < SIGDIM; ++s) acc += Wfin[t * SIGDIM + s] * red[s];
        logits[t] = bfin[t] + acc * (1.0f / (float)KPATH);
    }
    __syncthreads();
    if (t < NCLS) {
        float mx = logits[0];
#pragma unroll
        for (int j = 1; j < NCLS; ++j) mx = fmaxf(mx, logits[j]);
        float se = 0.0f;
#pragma unroll
        for (int j = 0; j < NCLS; ++j) se += expf(logits[j] - mx);
        out[b * NCLS + t] = logits[t] - mx - logf(se);
    }
}

// ---------------------------------------------------------------------------
// Launch
// ---------------------------------------------------------------------------
extern "C" void kernel_launch(void* const* d_in, const int* in_sizes, int n_in,
                              void* d_out, int out_size, void* d_ws, size_t ws_size,
                              hipStream_t stream) {
    const float* x     = (const float*)d_in[0];
    const float* Wmean = (const float*)d_in[1];
    const float* bmean = (const float*)d_in[2];
    const float* Wcov  = (const float*)d_in[3];
    const float* bcov  = (const float*)d_in[4];
    const float* Wfin  = (const float*)d_in[5];
    const float* bfin  = (const float*)d_in[6];
    const float* eps   = (const float*)d_in[7];
    const int*   xi    = (const int*)d_in[8];
    const int*   yi    = (const int*)d_in[9];
    float* out = (float*)d_out;

    // Workspace layout (floats); mean+Mbuf contiguous for one zero pass
    float* ws    = (float*)d_ws;
    float* xpad  = ws;                                  // 256*160    =   40960
    float* WmP   = xpad + BATCH * KPAD;                 // 96*160     =   15360
    float* WcP   = WmP  + MPAD * KPAD;                  // 2976*160   =  476160
    float* cov   = WcP  + MATEL * KPAD;                 // 256*2976   =  761856
    float* epsT  = cov  + BATCH * MATEL;                // 256*64*96  = 1572864
    float* newV  = epsT + BATCH * KPATH * MPAD;         // 256*96*64  = 1572864
    float* mean  = newV + BATCH * MPAD * KPATH;         // 256*96     =   24576
    float* Mbuf  = mean + BATCH * MPAD;                 // 256*96*96  = 2359296

    // 1. pads + zeroing (mean and Mbuf are contiguous: one grid-stride pass)
    pad_rows_k<<<(BATCH * KPAD) / 256, 256, 0, stream>>>(x, xpad, BATCH, BATCH);
    pad_rows_k<<<(MPAD * KPAD) / 256, 256, 0, stream>>>(Wmean, WmP, MDIM, MPAD);
    pad_rows_k<<<(MATEL * KPAD) / 256, 256, 0, stream>>>(Wcov, WcP, MATEL, MATEL);
    pad_eps_T<<<(BATCH * KPATH * MPAD) / 256, 256, 0, stream>>>(eps, epsT);
    zero_kernel<<<1024, 256, 0, stream>>>(mean, BATCH * MPAD + BATCH * MPAD * MPAD);

    // 2. mean = x @ Wmean^T + b   (6 col-tiles -> 2 blocks x 4 waves, 16 row tiles)
    gemm_xwT<<<dim3(2, 16), 128, 0, stream>>>(xpad, WmP, bmean, mean, MDIM, MPAD, 6);
    // 3. cov  = x @ Wcov^T + b    (186 col-tiles -> 47 blocks, 16 row tiles)
    gemm_xwT<<<dim3(47, 16), 128, 0, stream>>>(xpad, WcP, bcov, cov, MATEL, MATEL, 186);
    // 4. scatter banded entries into M
    scatter_cov<<<(BATCH * MATEL) / 256, 256, 0, stream>>>(cov, xi, yi, Mbuf);
    // 5. newV = M @ eps + mean    (one block per batch, 6 waves x (16x64))
    gemm_batched<<<BATCH, 192, 0, stream>>>(Mbuf, epsT, mean, newV);
    // 6. paths -> signatures -> normalize -> classify
    sig_kernel<<<BATCH, 256, 0, stream>>>(x, newV, Wfin, bfin, out);
}